// LengthRegulator_76072460746984
// MI455X (gfx1250) — compile-verified
//
#include <hip/hip_runtime.h>
#include <cstdint>

// ---------------------------------------------------------------------------
// LengthRegulator for MI455X (gfx1250), fused single kernel.
// out[b, j, :] = x[b, i(j), :] with i(j) = first i s.t. cum[b,i] > j, else 0;
// lengths[b] = cum[b, t_x-1].
// Memory-bound gather (~134 MB -> ~5.8 us @ 23.3 TB/s). Each block:
//   1) async-DMA the 4 KB duration row into LDS (ASYNCcnt path),
//   2) block-local inclusive scan (4 elem/thread + 256-wide Hillis-Steele),
//   3) 8 waves x 4 consecutive output rows: binary-search source phoneme,
//      gather row global->LDS->global with async b128 DMA, caching the staged
//      row across consecutive frames with the same source, NT store hint.
// ---------------------------------------------------------------------------

#define TX 1024                 // t_x (fixed by the reference setup)
#define WAVES 8                 // 256 threads / wave32
#define ROWS_PER_WAVE 4
#define ROWS_PER_BLOCK (WAVES * ROWS_PER_WAVE)   // 32

// gfx1250 async LDS DMA. NOTE (CDNA5 ISA): the 24-bit offset is
// added to BOTH the LDS address and the global address.
#define ASYNC_LOAD_B128(ldsoff, gptr, off)                                     \
  asm volatile("global_load_async_to_lds_b128 %0, %1, off offset:" #off        \
               :: "v"(ldsoff), "v"(gptr) : "memory")

// Output is a 117 MB write-once stream: non-temporal store hint keeps the
// heavily reused x rows resident in cache.
#define ASYNC_STORE_B128_NT(gptr, ldsoff, off)                                 \
  asm volatile("global_store_async_from_lds_b128 %0, %1, off offset:" #off     \
               " th:TH_STORE_NT"                                               \
               :: "v"(gptr), "v"(ldsoff) : "memory")

__device__ __forceinline__ void wait_async0() {
  asm volatile("s_wait_asynccnt 0x0" ::: "memory");
}

// Low 32 bits of a generic pointer into the shared aperture are the LDS byte
// address (CDNA5 ISA).
__device__ __forceinline__ unsigned lds_off(const void* p) {
  return (unsigned)(uintptr_t)p;
}

__global__ __launch_bounds__(256)
void lr_fused_kernel(const float* __restrict__ x, const int* __restrict__ rc,
                     float* __restrict__ out, float* __restrict__ len_out,
                     int t_y, int d) {
  __shared__ int   cum_s[TX];        // durations, then in-place cumulative sums
  __shared__ int   partial[256];     // per-thread partial sums for the scan
  __shared__ float stage[WAVES][256]; // one gathered row per wave

  const int b   = blockIdx.y;
  const int tid = threadIdx.x;

  // ---- 1) async-stage the duration row: 256 lanes x 16 B = 4 KB ------------
  const int* rc_b = rc + b * TX;
  ASYNC_LOAD_B128(lds_off(&cum_s[tid * 4]), (const void*)(rc_b + tid * 4), 0);
  wait_async0();
  __syncthreads();

  // ---- 2) block-wide inclusive scan of 1024 durations ----------------------
  const int4 v = *(const int4*)&cum_s[tid * 4];
  const int lsum = v.x + v.y + v.z + v.w;
  partial[tid] = lsum;
  __syncthreads();
#pragma unroll
  for (int off = 1; off < 256; off <<= 1) {
    int add = (tid >= off) ? partial[tid - off] : 0;
    __syncthreads();
    partial[tid] += add;
    __syncthreads();
  }
  {
    const int excl = partial[tid] - lsum;   // exclusive prefix for this chunk
    int4 cv;
    cv.x = excl + v.x;
    cv.y = cv.x + v.y;
    cv.z = cv.y + v.z;
    cv.w = cv.z + v.w;
    *(int4*)&cum_s[tid * 4] = cv;           // cum_s now holds inclusive cumsum
  }
  __syncthreads();

  const int total = partial[255];           // == cum_s[TX-1]
  if (blockIdx.x == 0 && tid == 0) len_out[b] = (float)total;

  // ---- 3) expansion: wave w handles 4 consecutive output frames ------------
  const int wave = tid >> 5;
  const int lane = tid & 31;
  const int j0 = blockIdx.x * ROWS_PER_BLOCK + wave * ROWS_PER_WAVE;

  const float* xb = x + (size_t)b * TX * (size_t)d;   // d == 256 per setup
  const unsigned lstage = lds_off(&stage[wave][lane * 8]);
  int prev_idx = -1;

  for (int r = 0; r < ROWS_PER_WAVE; ++r) {
    const int j = j0 + r;
    if (j >= t_y) break;                               // wave-uniform
    float* dst = out + ((size_t)b * t_y + j) * (size_t)d;

    if (j < total) {
      // Branchless lower-bound: idx = #(cum <= j) = first i with cum[i] > j.
      int idx = 0;
#pragma unroll
      for (int step = TX >> 1; step > 0; step >>= 1) {
        const int probe = idx + step;
        if (probe <= TX && cum_s[probe - 1] <= j) idx = probe;
      }
      if (idx != prev_idx) {                           // wave-uniform
        const float* g = xb + (size_t)idx * d + lane * 8;
        wait_async0();          // pending stores still read the stage buffer
        ASYNC_LOAD_B128(lstage, (const void*)g, 0);
        ASYNC_LOAD_B128(lstage, (const void*)g, 16);
        wait_async0();          // loads landed in LDS
        prev_idx = idx;
      }
      // Re-issue stores straight from the cached stage (no reload needed when
      // consecutive frames expand the same phoneme).
      float* gd = dst + lane * 8;
      ASYNC_STORE_B128_NT((void*)gd, lstage, 0);
      ASYNC_STORE_B128_NT((void*)gd, lstage, 16);
    } else {
      // Past this batch's total length: reference mask is all-zero here.
      const float4 z = make_float4(0.f, 0.f, 0.f, 0.f);
      *(float4*)(dst + lane * 8)     = z;
      *(float4*)(dst + lane * 8 + 4) = z;
    }
  }
  // Outstanding async stores drain at S_ENDPGM (implicit wait-idle).
}

// ---------------------------------------------------------------------------
// Launcher
// ---------------------------------------------------------------------------
extern "C" void kernel_launch(void* const* d_in, const int* in_sizes, int n_in,
                              void* d_out, int out_size, void* d_ws, size_t ws_size,
                              hipStream_t stream) {
  const float* x  = (const float*)d_in[0];   // (b, t_x, d) fp32
  const int*   rc = (const int*)d_in[1];     // (b, t_x) int

  const int B   = 16;
  const int d   = in_sizes[0] / in_sizes[1];       // 256
  const int t_y = (out_size / B - 1) / d;          // 7168 (out_size = b*(t_y*d+1))

  float* out     = (float*)d_out;                  // (b, t_y, d)
  float* len_out = out + (size_t)B * t_y * d;      // b floats (lengths)

  dim3 grid((t_y + ROWS_PER_BLOCK - 1) / ROWS_PER_BLOCK, B);
  lr_fused_kernel<<<grid, dim3(256), 0, stream>>>(x, rc, out, len_out, t_y, d);
}